// DCGN_5239860101881
// MI455X (gfx1250) — compile-verified
//
#include <hip/hip_runtime.h>
#include <math.h>

// B=64, N=1080, F=512, P=3 -> S1=360; H1=784 -> S2=120; H2=28; cls 84->32->10
// Key algebraic facts (see reference):
//  - _get_adjacent returns a diagonal matrix; _propagate's einsum factors into
//    Asum[b] * (sum_over_batch h) so the big per-batch matmul collapses to ONE
//    small fp32 GEMM + per-batch scalar scale.
//  - h1 (64x360x784, 72MB) is never materialized: layer-2 recomputes it from
//    the 1.1MB M1 = Hsum1 @ prop1_W (L2-resident).
// All GEMMs use V_WMMA_F32_16X16X4_F32 (fp32-exact matrix core path).

typedef float v2f __attribute__((ext_vector_type(2)));
typedef float v8f __attribute__((ext_vector_type(8)));

__device__ __forceinline__ float gelu_erf(float v) {
    // torch F.gelu default: 0.5*x*(1+erf(x/sqrt(2)))
    return 0.5f * v * (1.0f + erff(v * 0.70710678118654752440f));
}

// ------------------------------ zero fill -----------------------------------
__global__ void k_zero(float* __restrict__ p, int n) {
    int i = blockIdx.x * blockDim.x + threadIdx.x;
    if (i < n) p[i] = 0.0f;
}

// ---------------- layer-1: attention-pool ss + node-conv + Asum -------------
// One block per (batch, window). F=512, 128 threads x 4 f-values.
__global__ void k_l1_pool(const float* __restrict__ x,
                          const float* __restrict__ nc1w,  // [3,512]
                          const float* __restrict__ g1w,   // [512]
                          const float* __restrict__ g1b,   // [1]
                          float* __restrict__ A1sum,       // [64]
                          float* __restrict__ H1sum) {     // [360,512]
    const int S = 360, F = 512, NN = 1080;
    int blk = blockIdx.x;
    int bb = blk / S, s = blk % S;
    int t  = threadIdx.x;
    const float* xrow = x + ((size_t)bb * NN + 3 * (size_t)s) * F;

    float x0[4], x1[4], x2[4];
    float d0 = 0.f, d1 = 0.f, d2 = 0.f;
#pragma unroll
    for (int j = 0; j < 4; ++j) {
        int f = t + 128 * j;
        float w = g1w[f];
        x0[j] = xrow[f];
        x1[j] = xrow[F + f];
        x2[j] = xrow[2 * F + f];
        d0 += x0[j] * w; d1 += x1[j] * w; d2 += x2[j] * w;
    }
    __shared__ float r0[128], r1[128], r2[128];
    __shared__ float satt[3];
    r0[t] = d0; r1[t] = d1; r2[t] = d2;
    __syncthreads();
    for (int st = 64; st > 0; st >>= 1) {
        if (t < st) { r0[t] += r0[t + st]; r1[t] += r1[t + st]; r2[t] += r2[t + st]; }
        __syncthreads();
    }
    if (t == 0) {
        float bias = g1b[0];
        float l0 = r0[0] + bias, l1 = r1[0] + bias, l2 = r2[0] + bias;
        float m  = fmaxf(l0, fmaxf(l1, l2));
        float e0 = expf(l0 - m), e1 = expf(l1 - m), e2 = expf(l2 - m);
        float inv = 1.0f / (e0 + e1 + e2);
        satt[0] = e0 * inv; satt[1] = e1 * inv; satt[2] = e2 * inv;
    }
    __syncthreads();
    float a0 = satt[0], a1 = satt[1], a2 = satt[2];
    float sp = 0.f;
#pragma unroll
    for (int j = 0; j < 4; ++j) {
        float pv = a0 * x0[j] + a1 * x1[j] + a2 * x2[j];
        sp += pv * pv;
    }
    r0[t] = sp;
    __syncthreads();
    for (int st = 64; st > 0; st >>= 1) {
        if (t < st) r0[t] += r0[t + st];
        __syncthreads();
    }
    if (t == 0) {
        float ss    = r0[0];
        float cosii = ss / (sqrtf(ss) * sqrtf(ss));
        float maskv = (2 * s <= S - 1) ? 1.0f : 0.0f;
        float d     = 1.0f + cosii * maskv;
        float dinv  = 1.0f / sqrtf(d);
        unsafeAtomicAdd(&A1sum[bb], dinv * d * dinv);
    }
#pragma unroll
    for (int j = 0; j < 4; ++j) {
        int f = t + 128 * j;
        float hval = x0[j] * nc1w[f] + x1[j] * nc1w[F + f] + x2[j] * nc1w[2 * F + f];
        unsafeAtomicAdd(&H1sum[s * F + f], hval);
    }
}

// ----------------------- fp32 WMMA GEMM (16x16 tile / wave) ------------------
// C[M,N] = A[M,K] @ B[K,N] (row-major). K % 4 == 0. Optional bias+gelu.
// Edge tiles: indices are CLAMPED (not zeroed) — garbage products land only in
// D rows >= M / cols >= N, which the bounds-checked store never writes. This
// keeps the K-loop 100% branch-free (EXEC stays full, loads unconditional).
__global__ void k_wmma_gemm(const float* __restrict__ A, const float* __restrict__ B,
                            float* __restrict__ C, int M, int N, int K,
                            const float* __restrict__ bias, int act) {
    int lane  = threadIdx.x & 31;
    int lhalf = lane >> 4;       // 0: lanes 0-15, 1: lanes 16-31
    int lm    = lane & 15;
    int tilesN = (N + 15) >> 4;
    int tilesM = (M + 15) >> 4;
    int wave = blockIdx.x * (blockDim.x >> 5) + (threadIdx.x >> 5);
    if (wave >= tilesM * tilesN) return;
    int row0 = (wave / tilesN) << 4;
    int col0 = (wave % tilesN) << 4;

    int ar = min(row0 + lm, M - 1);  // A: M across lanes 0-15 (dup in 16-31)
    int bc = min(col0 + lm, N - 1);  // B: N across lanes 0-15 (dup in 16-31)
    // A frag: v0 holds K=0 (lanes 0-15) / K=2 (lanes 16-31); v1 holds K=1/K=3.
    const float* ap = A + (size_t)ar * K + 2 * lhalf;          // 8B aligned (K even)
    const float* bp = B + (size_t)(2 * lhalf) * N + bc;
    v8f acc = {};
#pragma unroll 4
    for (int k = 0; k < K; k += 4) {
        v2f a = *(const v2f*)ap;     // global_load_b64
        v2f b;
        b.x = bp[0];
        b.y = bp[N];
        acc = __builtin_amdgcn_wmma_f32_16x16x4_f32(false, a, false, b,
                                                    (short)0, acc, false, false);
        ap += 4;
        bp += (size_t)4 * N;
    }
#pragma unroll
    for (int r = 0; r < 8; ++r) {   // C/D: VGPR r -> M=r (lanes 0-15), M=r+8 (16-31)
        int rr = row0 + r + 8 * lhalf;
        int cc = col0 + lm;
        if (rr < M && cc < N) {
            float v = acc[r];
            if (bias) v += bias[cc];
            if (act)  v = gelu_erf(v);
            C[(size_t)rr * N + cc] = v;
        }
    }
}

// -------- layer-2: rebuild h1 on the fly, pool ss + node-conv + Asum --------
// One block (256 thr) per (batch, window). F2=784.
__global__ void k_l2_pool(const float* __restrict__ M1,     // [360,784]
                          const float* __restrict__ A1sum,  // [64]
                          const float* __restrict__ p1b,    // [784]
                          const float* __restrict__ nc2w,   // [3,784]
                          const float* __restrict__ g2w,    // [784]
                          const float* __restrict__ g2b,    // [1]
                          float* __restrict__ A2sum,        // [64]
                          float* __restrict__ H2sum) {      // [120,784]
    const int S = 120, F = 784;
    int blk = blockIdx.x;
    int bb = blk / S, s = blk % S;
    int t  = threadIdx.x;
    float a1 = A1sum[bb];

    float h0[4], h1v[4], h2v[4];
    float d0 = 0.f, d1 = 0.f, d2 = 0.f;
#pragma unroll
    for (int j = 0; j < 4; ++j) {
        h0[j] = h1v[j] = h2v[j] = 0.f;
        int f = t + 256 * j;
        if (f < F) {
            float pb = p1b[f];
            h0[j]  = gelu_erf(a1 * M1[(3 * s + 0) * F + f] + pb);
            h1v[j] = gelu_erf(a1 * M1[(3 * s + 1) * F + f] + pb);
            h2v[j] = gelu_erf(a1 * M1[(3 * s + 2) * F + f] + pb);
            float w = g2w[f];
            d0 += h0[j] * w; d1 += h1v[j] * w; d2 += h2v[j] * w;
        }
    }
    __shared__ float r0[256], r1[256], r2[256];
    __shared__ float satt[3];
    r0[t] = d0; r1[t] = d1; r2[t] = d2;
    __syncthreads();
    for (int st = 128; st > 0; st >>= 1) {
        if (t < st) { r0[t] += r0[t + st]; r1[t] += r1[t + st]; r2[t] += r2[t + st]; }
        __syncthreads();
    }
    if (t == 0) {
        float bias = g2b[0];
        float l0 = r0[0] + bias, l1 = r1[0] + bias, l2 = r2[0] + bias;
        float m  = fmaxf(l0, fmaxf(l1, l2));
        float e0 = expf(l0 - m), e1 = expf(l1 - m), e2 = expf(l2 - m);
        float inv = 1.0f / (e0 + e1 + e2);
        satt[0] = e0 * inv; satt[1] = e1 * inv; satt[2] = e2 * inv;
    }
    __syncthreads();
    float a0 = satt[0], aa1 = satt[1], a2 = satt[2];
    float sp = 0.f;
#pragma unroll
    for (int j = 0; j < 4; ++j) {
        float pv = a0 * h0[j] + aa1 * h1v[j] + a2 * h2v[j];
        sp += pv * pv;   // padded lanes contribute 0
    }
    r0[t] = sp;
    __syncthreads();
    for (int st = 128; st > 0; st >>= 1) {
        if (t < st) r0[t] += r0[t + st];
        __syncthreads();
    }
    if (t == 0) {
        float ss    = r0[0];
        float cosii = ss / (sqrtf(ss) * sqrtf(ss));
        float maskv = (2 * s <= S - 1) ? 1.0f : 0.0f;
        float d     = 1.0f + cosii * maskv;
        float dinv  = 1.0f / sqrtf(d);
        unsafeAtomicAdd(&A2sum[bb], dinv * d * dinv);
    }
#pragma unroll
    for (int j = 0; j < 4; ++j) {
        int f = t + 256 * j;
        if (f < F) {
            float hval = h0[j] * nc2w[f] + h1v[j] * nc2w[F + f] + h2v[j] * nc2w[2 * F + f];
            unsafeAtomicAdd(&H2sum[s * F + f], hval);
        }
    }
}

// ---- apply layer-2 scale+bias+gelu: X[b,s,c] = gelu(A2sum[b]*M2[s,c]+b[c]) --
__global__ void k_apply2(const float* __restrict__ M2,     // [120,28]
                         const float* __restrict__ A2sum,  // [64]
                         const float* __restrict__ p2b,    // [28]
                         float* __restrict__ X, int n) {   // [64*120*28] == [2560,84]
    int idx = blockIdx.x * blockDim.x + threadIdx.x;
    if (idx >= n) return;
    int c  = idx % 28;
    int s  = (idx / 28) % 120;
    int bb = idx / (28 * 120);
    X[idx] = gelu_erf(A2sum[bb] * M2[s * 28 + c] + p2b[c]);
}

// ------------------- final tiny GEMM: [2560,32]@[32,10]+b -------------------
__global__ void k_final(const float* __restrict__ HID,  // [2560,32]
                        const float* __restrict__ cw2,  // [32,10]
                        const float* __restrict__ cb2,  // [10]
                        float* __restrict__ out, int n) {
    int idx = blockIdx.x * blockDim.x + threadIdx.x;
    if (idx >= n) return;
    int row = idx / 10, c = idx % 10;
    float acc = cb2[c];
    const float* h = HID + row * 32;
#pragma unroll
    for (int j = 0; j < 32; ++j) acc += h[j] * cw2[j * 10 + c];
    out[idx] = acc;
}

extern "C" void kernel_launch(void* const* d_in, const int* in_sizes, int n_in,
                              void* d_out, int out_size, void* d_ws, size_t ws_size,
                              hipStream_t stream) {
    (void)in_sizes; (void)n_in; (void)out_size; (void)ws_size;
    const float* x    = (const float*)d_in[0];
    const float* nc1w = (const float*)d_in[1];
    const float* g1w  = (const float*)d_in[2];
    const float* g1b  = (const float*)d_in[3];
    const float* p1W  = (const float*)d_in[4];
    const float* p1b  = (const float*)d_in[5];
    const float* nc2w = (const float*)d_in[6];
    const float* g2w  = (const float*)d_in[7];
    const float* g2b  = (const float*)d_in[8];
    const float* p2W  = (const float*)d_in[9];
    const float* p2b  = (const float*)d_in[10];
    const float* cw1  = (const float*)d_in[11];
    const float* cb1  = (const float*)d_in[12];
    const float* cw2  = (const float*)d_in[13];
    const float* cb2  = (const float*)d_in[14];
    float* out = (float*)d_out;

    // workspace layout (floats); total 861,088 floats ~= 3.44 MB
    float* ws    = (float*)d_ws;
    float* A1sum = ws;              // 64
    float* A2sum = ws + 64;         // 64
    float* H1sum = ws + 128;        // 360*512 = 184320
    float* M1    = ws + 184448;     // 360*784 = 282240
    float* H2sum = ws + 466688;     // 120*784 = 94080
    float* M2    = ws + 560768;     // 120*28  = 3360
    float* X     = ws + 564128;     // 2560*84 = 215040
    float* HID   = ws + 779168;     // 2560*32 = 81920

    // zero all atomic accumulators (A1sum..H2sum range) every call
    k_zero<<<(560768 + 255) / 256, 256, 0, stream>>>(ws, 560768);

    // layer 1 fused pool/conv/adjacency over x (single pass over 141.6 MB)
    k_l1_pool<<<64 * 360, 128, 0, stream>>>(x, nc1w, g1w, g1b, A1sum, H1sum);

    // M1 = H1sum[360,512] @ prop1_W[512,784]  (fp32 WMMA)
    {
        int M = 360, N = 784, K = 512;
        int tiles = ((M + 15) / 16) * ((N + 15) / 16);
        k_wmma_gemm<<<(tiles + 7) / 8, 256, 0, stream>>>(H1sum, p1W, M1, M, N, K, nullptr, 0);
    }

    // layer 2 fused (h1 rebuilt on the fly from M1)
    k_l2_pool<<<64 * 120, 256, 0, stream>>>(M1, A1sum, p1b, nc2w, g2w, g2b, A2sum, H2sum);

    // M2 = H2sum[120,784] @ prop2_W[784,28]  (fp32 WMMA)
    {
        int M = 120, N = 28, K = 784;
        int tiles = ((M + 15) / 16) * ((N + 15) / 16);
        k_wmma_gemm<<<(tiles + 7) / 8, 256, 0, stream>>>(H2sum, p2W, M2, M, N, K, nullptr, 0);
    }

    // X = gelu(A2sum[b]*M2 + b2) flattened to [2560,84]
    k_apply2<<<(215040 + 255) / 256, 256, 0, stream>>>(M2, A2sum, p2b, X, 215040);

    // HID = gelu(X[2560,84] @ cls_w1[84,32] + cb1)  (fp32 WMMA, fused bias+gelu)
    {
        int M = 2560, N = 32, K = 84;
        int tiles = ((M + 15) / 16) * ((N + 15) / 16);
        k_wmma_gemm<<<(tiles + 7) / 8, 256, 0, stream>>>(X, cw1, HID, M, N, K, cb1, 1);
    }

    // out = HID @ cls_w2 + cb2
    k_final<<<(25600 + 255) / 256, 256, 0, stream>>>(HID, cw2, cb2, out, 25600);
}